// DetectionLoss_34849364640371
// MI455X (gfx1250) — compile-verified
//
#include <hip/hip_runtime.h>
#include <hip/hip_bf16.h>

#define BB 16
#define NN 64
#define HH 256
#define WW 256

// Workspace layout (floats):
//   [0, 4096)      per-box params float4 {cx, cy, floor(3*sigma), 1/(2*sigma^2)}
//   [4096, 4160)   per-image accumulators: {pos_loss, neg_loss, num_pos, num_neg} x B
//   [4160, 4164)   per-block box-L1 partial sums (4 prep blocks)
#define WS_PARAMS 0
#define WS_ACC    (BB * NN * 4)
#define WS_BOXP   (WS_ACC + BB * 4)

typedef __attribute__((ext_vector_type(4))) int v4i;
typedef __attribute__((address_space(1))) v4i* global_v4i_ptr;
typedef __attribute__((address_space(3))) v4i* lds_v4i_ptr;

__global__ void DetLoss_zero_kernel(float* __restrict__ acc) {
    int i = threadIdx.x;
    if (i < BB * 4) acc[i] = 0.0f;
}

__global__ __launch_bounds__(256) void DetLoss_prep_kernel(
        const float* __restrict__ tgt,      // [B,N,4]
        const float* __restrict__ pboxes,   // [B,2,H,W]
        float* __restrict__ ws_params,      // [B*N] float4
        float* __restrict__ ws_boxp) {      // [gridDim.x]
    int i = blockIdx.x * 256 + threadIdx.x;     // 0 .. B*N-1
    int b = i >> 6;
    float4 t = ((const float4*)tgt)[i];         // x0,y0,x1,y1
    float cxf = (t.x + t.z) * 0.5f * (float)WW;
    float cyf = (t.y + t.w) * 0.5f * (float)HH;
    float sw  = (t.z - t.x) * (float)WW;
    float sh  = (t.w - t.y) * (float)HH;
    int cx = (int)fminf(fmaxf(cxf, 0.0f), (float)(WW - 1));  // trunc == astype(int32), value >= 0
    int cy = (int)fminf(fmaxf(cyf, 0.0f), (float)(HH - 1));
    float sigma  = fmaxf(1.0f, 0.15f * (sw + sh) * 0.5f);
    float tmp    = floorf(3.0f * sigma);
    float inv2s2 = 1.0f / (2.0f * sigma * sigma);
    float4 o;
    o.x = (float)cx; o.y = (float)cy; o.z = tmp; o.w = inv2s2;
    ((float4*)ws_params)[i] = o;

    // box L1 at integer centers
    float tw  = t.z - t.x;
    float th  = t.w - t.y;
    float pb0 = pboxes[((size_t)(b * 2 + 0) * HH + cy) * WW + cx];
    float pb1 = pboxes[((size_t)(b * 2 + 1) * HH + cy) * WW + cx];
    float l1  = fabsf(pb0 - tw) + fabsf(pb1 - th);

    // wave32 + block reduction
    for (int off = 16; off > 0; off >>= 1) l1 += __shfl_xor(l1, off, 32);
    __shared__ float sred[8];
    if ((threadIdx.x & 31) == 0) sred[threadIdx.x >> 5] = l1;
    __syncthreads();
    if (threadIdx.x == 0) {
        float s = 0.0f;
        for (int w = 0; w < 8; ++w) s += sred[w];
        ws_boxp[blockIdx.x] = s;
    }
}

__global__ __launch_bounds__(256) void DetLoss_focal_kernel(
        const float* __restrict__ logits,   // [B,1,H,W]
        const float* __restrict__ ws_params,
        float* __restrict__ ws_acc) {
    __shared__ float4 sbox[NN];   // staged per-image box params
    __shared__ float4 ssel[NN];   // row-culled boxes {cx, dy*dy, tmp, inv2s2}
    __shared__ int   s_n;
    __shared__ float sacc[4];

    const int tid = threadIdx.x;              // == x (W == 256)
    const int b   = blockIdx.x >> 8;
    const int y   = blockIdx.x & (HH - 1);

    if (tid == 0) s_n = 0;
    if (tid < 4)  sacc[tid] = 0.0f;

    // ---- CDNA5 async global->LDS staging of the 64 box params (1 KB) ----
#if defined(__gfx1250__) && \
    __has_builtin(__builtin_amdgcn_global_load_async_to_lds_b128) && \
    __has_builtin(__builtin_amdgcn_s_wait_asynccnt)
    if (tid < NN) {
        const float4* g = ((const float4*)ws_params) + (b * NN + tid);
        __builtin_amdgcn_global_load_async_to_lds_b128(
            (global_v4i_ptr)g,
            (lds_v4i_ptr)&sbox[tid],
            /*offset=*/0, /*cpol=*/0);
    }
    __builtin_amdgcn_s_wait_asynccnt(0);
#else
    if (tid < NN) sbox[tid] = ((const float4*)ws_params)[b * NN + tid];
#endif
    __syncthreads();

    const float yf = (float)y;
    const float xf = (float)tid;

    // row culling: keep boxes with |y - cy| <= floor(3*sigma); fold dy^2 in
    if (tid < NN) {
        float4 bp = sbox[tid];
        float dy = yf - bp.y;
        if (fabsf(dy) <= bp.z) {
            int p = atomicAdd(&s_n, 1);
            ssel[p] = make_float4(bp.x, dy * dy, bp.z, bp.w);
        }
    }
    __syncthreads();

    const int nsel = s_n;
    float gt = 0.0f;
    for (int i = 0; i < nsel; ++i) {
        float4 s = ssel[i];
        float dx = xf - s.x;
        if (fabsf(dx) <= s.z) {
            float g = expf(-(dx * dx + s.y) * s.w);   // exp(0)==1.0 exactly at center
            gt = fmaxf(gt, g);
        }
    }

    // focal loss terms
    float z  = logits[((size_t)b * HH + y) * WW + tid];
    float p  = 1.0f / (1.0f + expf(-z));
    bool pos = (gt == 1.0f);
    float om = 1.0f - p;
    float pl = pos ? (-(om * om) * logf(fmaxf(p, 1e-6f))) : 0.0f;
    float tg = 1.0f - gt;
    float t2 = tg * tg;
    float nl = pos ? 0.0f : (-(t2 * t2) * (p * p) * logf(fmaxf(om, 1e-6f)));
    float np = pos ? 1.0f : 0.0f;
    float nn = 1.0f - np;

    // wave32 reduction, then shared atomics, then 4 global atomics per block
    for (int off = 16; off > 0; off >>= 1) {
        pl += __shfl_xor(pl, off, 32);
        nl += __shfl_xor(nl, off, 32);
        np += __shfl_xor(np, off, 32);
        nn += __shfl_xor(nn, off, 32);
    }
    if ((tid & 31) == 0) {
        atomicAdd(&sacc[0], pl);
        atomicAdd(&sacc[1], nl);
        atomicAdd(&sacc[2], np);
        atomicAdd(&sacc[3], nn);
    }
    __syncthreads();
    if (tid < 4) atomicAdd(&ws_acc[b * 4 + tid], sacc[tid]);
}

__global__ void DetLoss_final_kernel(const float* __restrict__ ws_acc,
                                     const float* __restrict__ ws_boxp,
                                     float* __restrict__ out) {
    if (threadIdx.x == 0) {
        float cls = 0.0f;
        for (int b = 0; b < BB; ++b) {
            float pl = ws_acc[b * 4 + 0];
            float nl = ws_acc[b * 4 + 1];
            float np = fmaxf(ws_acc[b * 4 + 2], 1.0f);
            float nn = fmaxf(ws_acc[b * 4 + 3], 1.0f);
            cls += pl / np + nl / nn;
        }
        cls *= (1.0f / (float)BB);
        float bs = ws_boxp[0] + ws_boxp[1] + ws_boxp[2] + ws_boxp[3];
        out[0] = cls;
        out[1] = bs / (float)(BB * NN * 2);
    }
}

extern "C" void kernel_launch(void* const* d_in, const int* in_sizes, int n_in,
                              void* d_out, int out_size, void* d_ws, size_t ws_size,
                              hipStream_t stream) {
    (void)in_sizes; (void)n_in; (void)out_size; (void)ws_size;
    const float* pred_logits = (const float*)d_in[0];   // [B,1,H,W]
    const float* pred_boxes  = (const float*)d_in[1];   // [B,2,H,W]
    const float* tgt_boxes   = (const float*)d_in[2];   // [B,N,4]
    float* out = (float*)d_out;
    float* ws  = (float*)d_ws;

    float* ws_params = ws + WS_PARAMS;
    float* ws_acc    = ws + WS_ACC;
    float* ws_boxp   = ws + WS_BOXP;

    hipLaunchKernelGGL(DetLoss_zero_kernel, dim3(1), dim3(64), 0, stream, ws_acc);
    hipLaunchKernelGGL(DetLoss_prep_kernel, dim3((BB * NN) / 256), dim3(256), 0, stream,
                       tgt_boxes, pred_boxes, ws_params, ws_boxp);
    hipLaunchKernelGGL(DetLoss_focal_kernel, dim3(BB * HH), dim3(256), 0, stream,
                       pred_logits, ws_params, ws_acc);
    hipLaunchKernelGGL(DetLoss_final_kernel, dim3(1), dim3(32), 0, stream,
                       ws_acc, ws_boxp, out);
}